// PAM_45801531245188
// MI455X (gfx1250) — compile-verified
//
#include <hip/hip_runtime.h>

// MI455X (gfx1250, wave32). Bandwidth-bound pipeline: 3 reads + 1 write of the
// 205MB tensor (~35us roofline at 23.3 TB/s). Tiny FC gates done with
// V_WMMA_F32_16X16X4_F32 (one wave per 16x16 tile, K=512).

#define BB      32
#define CC      512
#define HH      56
#define WW      56
#define HW      3136                 // 56*56
#define CHW     (CC * HW)
#define NPLANE  (BB * CC)            // 16384
#define NCHUNK  13                   // ceil(3136/256)

typedef __attribute__((ext_vector_type(2))) float v2f;
typedef __attribute__((ext_vector_type(8))) float v8f;

__device__ __forceinline__ float wred_sum(float v) {
#pragma unroll
  for (int o = 16; o > 0; o >>= 1) v += __shfl_xor(v, o, 32);
  return v;
}
__device__ __forceinline__ float wred_max(float v) {
#pragma unroll
  for (int o = 16; o > 0; o >>= 1) v = fmaxf(v, __shfl_xor(v, o, 32));
  return v;
}

// ---------------------------------------------------------------------------
// K1: per-(b,c) spatial sum/max of relu(x) -> keep flag (peak-avg >= 0.5).
// One block per plane; float4 coalesced reads.
// ---------------------------------------------------------------------------
__global__ __launch_bounds__(256) void k1_keep(const float* __restrict__ x,
                                               float* __restrict__ keep) {
  const int p = blockIdx.x;
  const float4* plane = (const float4*)(x + (size_t)p * HW);
  float s = 0.f, m = 0.f;  // relu(x) >= 0, so 0 is a valid max identity
  for (int i = threadIdx.x; i < HW / 4; i += 256) {
    float4 v = plane[i];
    float a = fmaxf(v.x, 0.f), b = fmaxf(v.y, 0.f);
    float c = fmaxf(v.z, 0.f), d = fmaxf(v.w, 0.f);
    s += (a + b) + (c + d);
    m = fmaxf(fmaxf(m, fmaxf(a, b)), fmaxf(c, d));
  }
  s = wred_sum(s);
  m = wred_max(m);
  __shared__ float ss[8], sm[8];
  if ((threadIdx.x & 31) == 0) { ss[threadIdx.x >> 5] = s; sm[threadIdx.x >> 5] = m; }
  __syncthreads();
  if (threadIdx.x == 0) {
    float st = ss[0], mt = sm[0];
    for (int w = 1; w < 8; ++w) { st += ss[w]; mt = fmaxf(mt, sm[w]); }
    keep[p] = ((mt - st * (1.f / HW)) < 0.5f) ? 0.f : 1.f;
  }
}

// ---------------------------------------------------------------------------
// K2: one block per (b, 256-wide hw chunk). Phase 1: channel max/mean over
// C=512 (coalesced: lane = hw) -> bb[b,hw]. Phase 2: re-read tile (L2-hot),
// z = bb*keep*relu(x), wave shuffle-reduce over hw, wave partials in LDS,
// write deterministic per-chunk partial max/sum per channel.
// ---------------------------------------------------------------------------
__global__ __launch_bounds__(256) void k2_bb_partials(
    const float* __restrict__ x, const float* __restrict__ keep,
    float* __restrict__ bb_out,
    float* __restrict__ pmax_part, float* __restrict__ psum_part) {
  const int b = blockIdx.y, chunk = blockIdx.x;
  const int tid = threadIdx.x;
  const int hw = chunk * 256 + tid;
  const bool valid = hw < HW;

  __shared__ float keep_s[CC];
  keep_s[tid]       = keep[b * CC + tid];
  keep_s[tid + 256] = keep[b * CC + tid + 256];
  __syncthreads();

  const float* xb = x + (size_t)b * CHW;
  float cmax = 0.f, csum = 0.f;
  if (valid) {
    for (int c = 0; c < CC; ++c) {
      float y = fmaxf(xb[(size_t)c * HW + hw], 0.f) * keep_s[c];
      cmax = fmaxf(cmax, y);
      csum += y;
    }
  }
  const float bbv = 1.f - __expf(-fabsf(cmax - csum * (1.f / CC)));
  if (valid) bb_out[b * HW + hw] = bbv;

  __shared__ float wmax_s[8 * CC];  // 16 KB
  __shared__ float wsum_s[8 * CC];  // 16 KB
  const int wv = tid >> 5, lane = tid & 31;
  for (int c = 0; c < CC; ++c) {
    float z = 0.f;
    if (valid) z = bbv * fmaxf(xb[(size_t)c * HW + hw], 0.f) * keep_s[c];
    float zm = wred_max(z);
    float zs = wred_sum(z);
    if (lane == 0) { wmax_s[wv * CC + c] = zm; wsum_s[wv * CC + c] = zs; }
  }
  __syncthreads();
  for (int c = tid; c < CC; c += 256) {
    float m = 0.f, s = 0.f;
#pragma unroll
    for (int w = 0; w < 8; ++w) {
      m = fmaxf(m, wmax_s[w * CC + c]);
      s += wsum_s[w * CC + c];
    }
    pmax_part[(size_t)chunk * NPLANE + b * CC + c] = m;
    psum_part[(size_t)chunk * NPLANE + b * CC + c] = s;
  }
}

// ---------------------------------------------------------------------------
// K2b: fixed-order combine of the 13 chunk partials -> xmax[b,c], psum[b,c].
// (No float atomics anywhere -> bitwise deterministic.)
// ---------------------------------------------------------------------------
__global__ __launch_bounds__(256) void k2b_combine(
    const float* __restrict__ pmax_part, const float* __restrict__ psum_part,
    float* __restrict__ xmax, float* __restrict__ psum) {
  const int p = blockIdx.x * 256 + threadIdx.x;
  if (p >= NPLANE) return;
  float m = 0.f, s = 0.f;
#pragma unroll
  for (int ch = 0; ch < NCHUNK; ++ch) {
    m = fmaxf(m, pmax_part[(size_t)ch * NPLANE + p]);
    s += psum_part[(size_t)ch * NPLANE + p];
  }
  xmax[p] = m;
  psum[p] = s;
}

// ---------------------------------------------------------------------------
// K3: control gates via V_WMMA_F32_16X16X4_F32.
// control[b][c] = act( sum_k pooled[b][k] * W[c][k] ), pooled = psum/3136.
// One wave per 16x16 output tile; blockIdx = (ntile, mtile, which_W).
// A frag (16x4 f32): lanes 0-15 K=k,k+1 ; lanes 16-31 K=k+2,k+3.
// B frag (4x16)  : N across lanes, same K split.
// D: VGPR r -> row r (lanes 0-15) / row r+8 (lanes 16-31), N = lane&15.
// ---------------------------------------------------------------------------
__global__ __launch_bounds__(32) void k3_gemm(
    const float* __restrict__ psum,
    const float* __restrict__ W1, const float* __restrict__ W3,
    float* __restrict__ c1, float* __restrict__ c3) {
  const int nt  = blockIdx.x;   // 0..31  (N tile of 16 channels)
  const int mt  = blockIdx.y;   // 0..1   (M tile of 16 batch rows)
  const int mat = blockIdx.z;   // 0: W1->sigmoid, 1: W3->relu/1.2
  const float* Wm = mat ? W3 : W1;
  float* outp     = mat ? c3 : c1;

  const int lane = threadIdx.x;
  const int half = lane >> 4;       // which K-pair this lane holds
  const int lrow = lane & 15;
  const float inv = 1.f / (float)HW;

  const float* arow = psum + (size_t)(mt * 16 + lrow) * CC;  // pooled row (un-normalized)
  const float* brow = Wm   + (size_t)(nt * 16 + lrow) * CC;  // W row n -> B column n

  v8f acc = {};
  for (int k = 0; k < CC; k += 4) {
    const int kb = k + half * 2;
    v2f a = *(const v2f*)(arow + kb);
    a = a * inv;                       // fold the mean normalization into A
    v2f bf = *(const v2f*)(brow + kb);
    acc = __builtin_amdgcn_wmma_f32_16x16x4_f32(
        /*neg_a=*/false, a, /*neg_b=*/false, bf,
        /*c_mod=*/(short)0, acc, /*reuse_a=*/false, /*reuse_b=*/false);
  }

#pragma unroll
  for (int r = 0; r < 8; ++r) {
    const int row = mt * 16 + r + half * 8;   // batch index
    const float d = acc[r];
    float o;
    if (mat == 0) {
      o = 1.f / (1.f + __expf(-d));           // sigmoid
    } else {
      o = fmaxf(d, 0.f);                      // relu
      o = (o < 1.f) ? 1.2f : o;               // values below 1 -> 1.2
    }
    outp[(size_t)row * CC + (nt * 16 + lrow)] = o;
  }
}

// ---------------------------------------------------------------------------
// K4: final elementwise pass. z = bb*keep*relu(x); out = (z < thres) ? 0 : z
// with thres = c1*c3*xmax (per-plane scalar). float4 read + write.
// ---------------------------------------------------------------------------
__global__ __launch_bounds__(256) void k4_final(
    const float* __restrict__ x, const float* __restrict__ keep,
    const float* __restrict__ bb, const float* __restrict__ xmax,
    const float* __restrict__ c1, const float* __restrict__ c3,
    float* __restrict__ out) {
  const int p = blockIdx.x;      // b*C + c
  const int b = p >> 9;          // / 512
  const float kp    = keep[p];
  const float thres = c1[p] * c3[p] * xmax[p];
  const float4* px = (const float4*)(x  + (size_t)p * HW);
  const float4* pb = (const float4*)(bb + (size_t)b * HW);
  float4*       po = (float4*)(out + (size_t)p * HW);
  for (int i = threadIdx.x; i < HW / 4; i += 256) {
    float4 v = px[i], w = pb[i], o;
    float z;
    z = w.x * fmaxf(v.x, 0.f) * kp; o.x = (z < thres) ? 0.f : z;
    z = w.y * fmaxf(v.y, 0.f) * kp; o.y = (z < thres) ? 0.f : z;
    z = w.z * fmaxf(v.z, 0.f) * kp; o.z = (z < thres) ? 0.f : z;
    z = w.w * fmaxf(v.w, 0.f) * kp; o.w = (z < thres) ? 0.f : z;
    po[i] = o;
  }
}

// ---------------------------------------------------------------------------
extern "C" void kernel_launch(void* const* d_in, const int* in_sizes, int n_in,
                              void* d_out, int out_size, void* d_ws, size_t ws_size,
                              hipStream_t stream) {
  const float* x  = (const float*)d_in[0];  // [32,512,56,56]
  const float* W1 = (const float*)d_in[1];  // [512,512]
  const float* W3 = (const float*)d_in[2];  // [512,512]
  float* out = (float*)d_out;

  // Workspace layout (floats); total ~2.4 MB.
  float* w         = (float*)d_ws;
  float* keep      = w;                 // 16384
  float* xmax      = w + 16384;         // 16384
  float* psum      = w + 32768;         // 16384
  float* c1        = w + 49152;         // 16384
  float* c3        = w + 65536;         // 16384
  float* bb        = w + 81920;         // 100352 (B*HW)
  float* pmax_part = w + 182272;        // 13*16384
  float* psum_part = w + 395264;        // 13*16384

  k1_keep<<<NPLANE, 256, 0, stream>>>(x, keep);
  k2_bb_partials<<<dim3(NCHUNK, BB), 256, 0, stream>>>(x, keep, bb, pmax_part, psum_part);
  k2b_combine<<<NPLANE / 256, 256, 0, stream>>>(pmax_part, psum_part, xmax, psum);
  k3_gemm<<<dim3(CC / 16, BB / 16, 2), 32, 0, stream>>>(psum, W1, W3, c1, c3);
  k4_final<<<NPLANE, 256, 0, stream>>>(x, keep, bb, xmax, c1, c3, out);
}